// ATT0_40707700032104
// MI455X (gfx1250) — compile-verified
//
#include <hip/hip_runtime.h>
#include <math.h>

// Problem constants from the reference: embeddings [B=512, I=128, D=64] f32.
#define B_  512
#define I_  128
#define D_  64
#define F4B (I_ * D_ / 4)   // 2048 float4 per batch (32 KB)
#define NT  256             // threads per block = 8 wave32
#define PT  (F4B / NT)      // 8 float4 per thread

// ---- gfx1250 hardware tanh (v_tanh_f32) ----
#if __has_builtin(__builtin_amdgcn_tanhf)
#define HW_TANH(x) __builtin_amdgcn_tanhf(x)
#elif __has_builtin(__builtin_amdgcn_tanh_f32)
#define HW_TANH(x) __builtin_amdgcn_tanh_f32(x)
#else
#define HW_TANH(x) tanhf(x)
#endif

// ---- async global -> LDS staging (ASYNCcnt path, codegen-confirmed) ----
#define USE_ASYNC_LDS __has_builtin(__builtin_amdgcn_global_load_async_to_lds_b128)
// ---- WMMA-based column-sum reduction (ones-vector GEMV on matrix pipe) ----
#define USE_WMMA __has_builtin(__builtin_amdgcn_wmma_f32_16x16x4_f32)

typedef int v4i __attribute__((ext_vector_type(4)));
typedef __attribute__((address_space(1))) v4i as1_v4i;
typedef __attribute__((address_space(3))) v4i as3_v4i;
typedef float v2f __attribute__((ext_vector_type(2)));
typedef float v8f __attribute__((ext_vector_type(8)));

__global__ __launch_bounds__(NT) void ctx_mean_tanh_kernel(
    const float* __restrict__ emb, float* __restrict__ out)
{
    __shared__ float4 tile[F4B];      // 32 KB staging for one batch
#if USE_WMMA
    __shared__ float redf[2 * D_];    // 512 B: per-half column sums
#else
    __shared__ float4 red[NT];        // 4 KB column-partial scratch (fallback)
#endif

    const int t = threadIdx.x;
    const int b = blockIdx.x;         // one workgroup per batch
    const float4* __restrict__ gsrc = (const float4*)emb + (size_t)b * F4B;
    float4* __restrict__ gdst       = (float4*)out       + (size_t)b * F4B;

    __builtin_prefetch((const void*)(gsrc + t), 0, 0);  // global_prefetch_b8

#if USE_ASYNC_LDS
    // Stage the batch tile through LDS with async b128 copies (ASYNCcnt).
#pragma unroll
    for (int k = 0; k < PT; ++k) {
        __builtin_amdgcn_global_load_async_to_lds_b128(
            (as1_v4i*)(gsrc + t + k * NT),
            (as3_v4i*)(tile + t + k * NT),
            /*offset=*/0, /*cpol=*/0);
    }
#if __has_builtin(__builtin_amdgcn_s_wait_asynccnt)
    __builtin_amdgcn_s_wait_asynccnt(0);
#else
    asm volatile("s_wait_asynccnt 0x0" ::: "memory");
#endif
#else
#pragma unroll
    for (int k = 0; k < PT; ++k) tile[t + k * NT] = gsrc[t + k * NT];
#endif
    __syncthreads();

    // Pull this thread's 8 float4s into registers (single HBM read total).
    float4 v[PT];
#pragma unroll
    for (int k = 0; k < PT; ++k) v[k] = tile[t + k * NT];

    const float inv = 1.0f / (float)I_;
    float mx, my, mz, mw;

#if USE_WMMA
    {
        // mean_i emb[i][d] as ones(1x128) . emb(128x64) on the matrix pipe.
        // Wave w: column chunk (w&3)*16..+15, row half (w>>2)*64..+63.
        // 16 chained V_WMMA_F32_16X16X4_F32 with A==1 accumulate
        // D[m][n] = sum_k B[k][n]  (replicated over m, permutation-invariant
        // in K, so the exact K<->slot layout of the B operand is irrelevant).
        const float* tf = (const float*)tile;
        const int lane   = t & 31;
        const int w      = t >> 5;
        const int cchunk = w & 3;
        const int h      = w >> 2;
        const int col    = cchunk * 16 + (lane & 15);
        const int roff   = (lane < 16) ? 0 : 2;   // K-rows carried by this half-wave
        const v2f a = {1.0f, 1.0f};               // ones A-matrix (16x4)
        v8f c = {};
#pragma unroll
        for (int s = 0; s < 16; ++s) {
            const int i0 = h * 64 + s * 4 + roff;
            v2f bb = { tf[i0 * D_ + col], tf[(i0 + 1) * D_ + col] };
            c = __builtin_amdgcn_wmma_f32_16x16x4_f32(
                /*neg_a=*/false, a, /*neg_b=*/false, bb,
                /*c_mod=*/(short)0, c, /*reuse_a=*/false, /*reuse_b=*/false);
        }
        // Lanes 0..15, VGPR0 (c[0]) hold the 64-row sum for column `col`.
        if (lane < 16) redf[h * D_ + col] = c[0];
        __syncthreads();
        const int d0 = 4 * (t & 15);   // this thread's d-columns 4c..4c+3
        mx = (redf[d0 + 0] + redf[D_ + d0 + 0]) * inv;
        my = (redf[d0 + 1] + redf[D_ + d0 + 1]) * inv;
        mz = (redf[d0 + 2] + redf[D_ + d0 + 2]) * inv;
        mw = (redf[d0 + 3] + redf[D_ + d0 + 3]) * inv;
    }
#else
    {
        // VALU fallback: two-level LDS reduction.
        float4 p = v[0];
#pragma unroll
        for (int k = 1; k < PT; ++k) {
            p.x += v[k].x; p.y += v[k].y; p.z += v[k].z; p.w += v[k].w;
        }
        red[t] = p;
        __syncthreads();
        const int cc = t & 15;
        float sx = 0.f, sy = 0.f, sz = 0.f, sw = 0.f;
#pragma unroll
        for (int j = 0; j < 16; ++j) {
            float4 r = red[cc + 16 * j];
            sx += r.x; sy += r.y; sz += r.z; sw += r.w;
        }
        mx = sx * inv; my = sy * inv; mz = sz * inv; mw = sw * inv;
    }
#endif

    // out = tanh(v * mean), values register-resident.
#pragma unroll
    for (int k = 0; k < PT; ++k) {
        float4 o;
        o.x = HW_TANH(v[k].x * mx);
        o.y = HW_TANH(v[k].y * my);
        o.z = HW_TANH(v[k].z * mz);
        o.w = HW_TANH(v[k].w * mw);
        gdst[t + k * NT] = o;
    }
}

extern "C" void kernel_launch(void* const* d_in, const int* in_sizes, int n_in,
                              void* d_out, int out_size, void* d_ws, size_t ws_size,
                              hipStream_t stream) {
    (void)in_sizes; (void)n_in; (void)out_size; (void)d_ws; (void)ws_size;
    const float* emb = (const float*)d_in[0];
    float* out = (float*)d_out;
    ctx_mean_tanh_kernel<<<dim3(B_), dim3(NT), 0, stream>>>(emb, out);
}